// RGAT_average_heads_76725295775761
// MI455X (gfx1250) — compile-verified
//
#include <hip/hip_runtime.h>
#include <hip/hip_bf16.h>

typedef __attribute__((ext_vector_type(2))) float v2f;
typedef __attribute__((ext_vector_type(8))) float v8f;

#define N_NODES 30000
#define N_EDGES 480000
#define N_REL   8
#define N_BASIS 4
#define N_HEADS 2
#define OUT_TOT 128   // H * out_ch == 128 for every layer
#define OUT_CH  64

// ---------------------------------------------------------------------------
// helpers
// ---------------------------------------------------------------------------
__device__ __forceinline__ void atomicMaxFloat(float* addr, float val) {
    // ordered-int trick: signed max for >=0, unsigned min for <0
    if (val >= 0.0f)
        atomicMax((int*)addr, __float_as_int(val));
    else
        atomicMin((unsigned int*)addr, (unsigned int)__float_as_int(val));
}

// ---------------------------------------------------------------------------
// 0) W[r] = sum_b att[r,b] * basis[b]        (R x in x 128, tiny)
// ---------------------------------------------------------------------------
__global__ void k_basis_combine(const float* __restrict__ basis,  // [B][in][128]
                                const float* __restrict__ att,    // [R][B]
                                float* __restrict__ W,            // [R][in][128]
                                int in_ch) {
    int total = N_REL * in_ch * OUT_TOT;
    int idx = blockIdx.x * blockDim.x + threadIdx.x;
    if (idx >= total) return;
    int r  = idx / (in_ch * OUT_TOT);
    int io = idx % (in_ch * OUT_TOT);
    float acc = 0.0f;
#pragma unroll
    for (int b = 0; b < N_BASIS; ++b)
        acc += att[r * N_BASIS + b] * basis[(size_t)b * in_ch * OUT_TOT + io];
    W[idx] = acc;
}

// ---------------------------------------------------------------------------
// init buffers: m = -inf, s = 0 (size N*H);  agg = 0 (size N*128)
// ---------------------------------------------------------------------------
__global__ void k_init_ms(float* __restrict__ m, float* __restrict__ s) {
    int i = blockIdx.x * blockDim.x + threadIdx.x;
    if (i < N_NODES * N_HEADS) { m[i] = -__builtin_huge_valf(); s[i] = 0.0f; }
}
__global__ void k_zero_agg(float* __restrict__ agg) {
    int i = blockIdx.x * blockDim.x + threadIdx.x;
    if (i < N_NODES * OUT_TOT) agg[i] = 0.0f;
}

// ---------------------------------------------------------------------------
// 1) xw[r] = X @ W[r]   via V_WMMA_F32_16X16X4_F32  (full f32 precision)
//    block = 256 threads = 8 waves; wave w owns output columns [16w,16w+16)
//    grid  = (N/16, R)
// ---------------------------------------------------------------------------
template <int IN_CH>
__global__ void k_gemm_xw(const float* __restrict__ X,   // [N][IN_CH]
                          const float* __restrict__ W,   // [R][IN_CH][128]
                          float* __restrict__ XW) {      // [R][N][128]
    __shared__ float lx[16][IN_CH + 4];                  // pitch IN_CH+4: conflict-free
    const int node0 = blockIdx.x * 16;
    const int r     = blockIdx.y;
    const int tid   = threadIdx.x;

    for (int i = tid; i < 16 * IN_CH; i += 256) {
        int row = i / IN_CH, col = i % IN_CH;
        lx[row][col] = X[(size_t)(node0 + row) * IN_CH + col];
    }
    __syncthreads();

    const int wave  = tid >> 5;
    const int lane  = tid & 31;
    const int half  = lane >> 4;     // 0: K=k0,k0+1   1: K=k0+2,k0+3
    const int l16   = lane & 15;
    const int ncol0 = wave * 16;
    const float* Wr = W + (size_t)r * IN_CH * OUT_TOT;

    v8f acc = {};
#pragma unroll 4
    for (int k0 = 0; k0 < IN_CH; k0 += 4) {
        const int ka = k0 + half * 2;
        v2f a, b;
        a.x = lx[l16][ka];
        a.y = lx[l16][ka + 1];
        b.x = Wr[(size_t)ka       * OUT_TOT + ncol0 + l16];
        b.y = Wr[(size_t)(ka + 1) * OUT_TOT + ncol0 + l16];
        acc = __builtin_amdgcn_wmma_f32_16x16x4_f32(false, a, false, b,
                                                    (short)0, acc, false, false);
    }

    float* out = XW + ((size_t)r * N_NODES + node0) * OUT_TOT + ncol0;
#pragma unroll
    for (int i = 0; i < 8; ++i) {
        int row = i + half * 8;                    // C/D: M=i (lanes 0-15), M=i+8 (16-31)
        out[(size_t)row * OUT_TOT + l16] = acc[i];
    }
}

// ---------------------------------------------------------------------------
// 2) edge logits: alpha = leaky_relu(out_i@q + out_j@k); atomic segment-max
//    one wave per edge (wave32)
// ---------------------------------------------------------------------------
__global__ void k_logits(const float* __restrict__ XW,
                         const long long* __restrict__ src,
                         const long long* __restrict__ dst,
                         const long long* __restrict__ etype,
                         const float* __restrict__ q,     // [128][2]
                         const float* __restrict__ kk,    // [128][2]
                         float* __restrict__ logits,      // [E][2]
                         float* __restrict__ mbuf) {      // [N][2]
    int e = blockIdx.x * 8 + (threadIdx.x >> 5);
    if (e >= N_EDGES) return;
    const int lane = threadIdx.x & 31;
    const long long sN = src[e], dN = dst[e];
    const int rt = (int)etype[e];
    const float* oi = XW + ((size_t)rt * N_NODES + (size_t)dN) * OUT_TOT;
    const float* oj = XW + ((size_t)rt * N_NODES + (size_t)sN) * OUT_TOT;

    float p0 = 0.0f, p1 = 0.0f;
#pragma unroll
    for (int j = 0; j < 4; ++j) {
        int c = lane + j * 32;
        float vi = oi[c], vj = oj[c];
        p0 += vi * q[c * 2 + 0] + vj * kk[c * 2 + 0];
        p1 += vi * q[c * 2 + 1] + vj * kk[c * 2 + 1];
    }
#pragma unroll
    for (int off = 16; off > 0; off >>= 1) {
        p0 += __shfl_down(p0, off, 32);
        p1 += __shfl_down(p1, off, 32);
    }
    if (lane == 0) {
        float l0 = p0 > 0.0f ? p0 : 0.2f * p0;
        float l1 = p1 > 0.0f ? p1 : 0.2f * p1;
        logits[e * 2 + 0] = l0;
        logits[e * 2 + 1] = l1;
        atomicMaxFloat(&mbuf[dN * 2 + 0], l0);
        atomicMaxFloat(&mbuf[dN * 2 + 1], l1);
    }
}

// ---------------------------------------------------------------------------
// 3) scatter: w = exp(logit - m[dst]); s[dst]+=w; agg[dst] += w * out_j
//    softmax denominator folded into the combine step
// ---------------------------------------------------------------------------
__global__ void k_scatter(const float* __restrict__ XW,
                          const long long* __restrict__ src,
                          const long long* __restrict__ dst,
                          const long long* __restrict__ etype,
                          const float* __restrict__ logits,
                          const float* __restrict__ mbuf,
                          float* __restrict__ sbuf,       // [N][2]
                          float* __restrict__ agg) {      // [N][128]
    int e = blockIdx.x * 8 + (threadIdx.x >> 5);
    if (e >= N_EDGES) return;
    const int lane = threadIdx.x & 31;
    const long long sN = src[e], dN = dst[e];
    const int rt = (int)etype[e];
    const float w0 = __expf(logits[e * 2 + 0] - mbuf[dN * 2 + 0]);
    const float w1 = __expf(logits[e * 2 + 1] - mbuf[dN * 2 + 1]);
    if (lane == 0) {
        atomicAdd(&sbuf[dN * 2 + 0], w0);
        atomicAdd(&sbuf[dN * 2 + 1], w1);
    }
    const float* oj = XW + ((size_t)rt * N_NODES + (size_t)sN) * OUT_TOT;
    float* ad = agg + (size_t)dN * OUT_TOT;
#pragma unroll
    for (int j = 0; j < 4; ++j) {
        int c = lane + j * 32;
        float wv = (c < 64) ? w0 : w1;
        atomicAdd(&ad[c], oj[c] * wv);
    }
}

// ---------------------------------------------------------------------------
// 4) combine: out = mean_h(agg/s) + bias + hin@skipW.T + skipb
// ---------------------------------------------------------------------------
__global__ void k_combine(const float* __restrict__ agg,
                          const float* __restrict__ sbuf,
                          const float* __restrict__ bias,   // [64]
                          const float* __restrict__ hin,    // [N][in_ch]
                          int in_ch,
                          const float* __restrict__ skipW,  // [64][in_ch]
                          const float* __restrict__ skipb,  // [64]
                          float* __restrict__ out) {        // [N][64]
    int idx = blockIdx.x * blockDim.x + threadIdx.x;
    if (idx >= N_NODES * OUT_CH) return;
    const int node = idx >> 6;
    const int c    = idx & 63;
    const float s0 = sbuf[node * 2 + 0] + 1e-16f;
    const float s1 = sbuf[node * 2 + 1] + 1e-16f;
    float v = 0.5f * (agg[(size_t)node * OUT_TOT + c] / s0 +
                      agg[(size_t)node * OUT_TOT + 64 + c] / s1) + bias[c];
    const float* hr = hin   + (size_t)node * in_ch;
    const float* wr = skipW + (size_t)c    * in_ch;
    float sk = 0.0f;
    for (int k = 0; k < in_ch; ++k) sk += hr[k] * wr[k];
    out[idx] = v + sk + skipb[c];
}

// ---------------------------------------------------------------------------
// 5) fused batchnorm + relu, in place: one block per channel (64 blocks)
// ---------------------------------------------------------------------------
__global__ void k_bn_relu(float* __restrict__ h,
                          const float* __restrict__ g,
                          const float* __restrict__ b) {
    const int c   = blockIdx.x;
    const int tid = threadIdx.x;
    __shared__ float ssum[256], ssq[256];
    float s = 0.0f, q = 0.0f;
    for (int n = tid; n < N_NODES; n += 256) {
        float v = h[(size_t)n * OUT_CH + c];
        s += v; q += v * v;
    }
    ssum[tid] = s; ssq[tid] = q;
    __syncthreads();
    for (int off = 128; off > 0; off >>= 1) {
        if (tid < off) { ssum[tid] += ssum[tid + off]; ssq[tid] += ssq[tid + off]; }
        __syncthreads();
    }
    const float mu  = ssum[0] * (1.0f / N_NODES);
    const float var = fmaxf(ssq[0] * (1.0f / N_NODES) - mu * mu, 0.0f);
    const float inv = rsqrtf(var + 1e-5f) * g[c];
    const float bb  = b[c];
    for (int n = tid; n < N_NODES; n += 256) {
        size_t i = (size_t)n * OUT_CH + c;
        float v = (h[i] - mu) * inv + bb;
        h[i] = v > 0.0f ? v : 0.0f;
    }
}

// ---------------------------------------------------------------------------
// host orchestration
// ---------------------------------------------------------------------------
extern "C" void kernel_launch(void* const* d_in, const int* in_sizes, int n_in,
                              void* d_out, int out_size, void* d_ws, size_t ws_size,
                              hipStream_t stream) {
    (void)in_sizes; (void)n_in; (void)out_size; (void)ws_size;

    const float*     x     = (const float*)d_in[0];
    const long long* eidx  = (const long long*)d_in[1];   // [2][E] int64
    const long long* etype = (const long long*)d_in[2];   // [E]    int64
    const long long* src   = eidx;
    const long long* dst   = eidx + N_EDGES;

    // workspace carve-up (floats)
    float* W      = (float*)d_ws;                         //  8*128*128
    float* xw     = W      + (size_t)N_REL * 128 * OUT_TOT;  // 8*30000*128
    float* logits = xw     + (size_t)N_REL * N_NODES * OUT_TOT;
    float* mbuf   = logits + (size_t)N_EDGES * N_HEADS;
    float* sbuf   = mbuf   + (size_t)N_NODES * N_HEADS;
    float* agg    = sbuf   + (size_t)N_NODES * N_HEADS;
    float* hA     = agg    + (size_t)N_NODES * OUT_TOT;
    float* hB     = hA     + (size_t)N_NODES * OUT_CH;

    struct Layer {
        const float *basis, *att, *q, *k, *bias, *skipW, *skipb, *bng, *bnb;
        const float* hin; float* hout; int in_ch; bool bn;
    };
    Layer L[3];
    L[0] = { (const float*)d_in[3],  (const float*)d_in[4],  (const float*)d_in[5],
             (const float*)d_in[6],  (const float*)d_in[7],  (const float*)d_in[8],
             (const float*)d_in[9],  (const float*)d_in[10], (const float*)d_in[11],
             x,  hA, 128, true };
    L[1] = { (const float*)d_in[12], (const float*)d_in[13], (const float*)d_in[14],
             (const float*)d_in[15], (const float*)d_in[16], (const float*)d_in[17],
             (const float*)d_in[18], (const float*)d_in[19], (const float*)d_in[20],
             hA, hB, 64, true };
    L[2] = { (const float*)d_in[21], (const float*)d_in[22], (const float*)d_in[23],
             (const float*)d_in[24], (const float*)d_in[25], (const float*)d_in[26],
             (const float*)d_in[27], nullptr, nullptr,
             hB, (float*)d_out, 64, false };

    for (int l = 0; l < 3; ++l) {
        const Layer& P = L[l];

        {   // W_r = sum_b att * basis
            int total = N_REL * P.in_ch * OUT_TOT;
            k_basis_combine<<<(total + 255) / 256, 256, 0, stream>>>(
                P.basis, P.att, W, P.in_ch);
        }
        k_init_ms<<<(N_NODES * N_HEADS + 255) / 256, 256, 0, stream>>>(mbuf, sbuf);
        k_zero_agg<<<(N_NODES * OUT_TOT + 255) / 256, 256, 0, stream>>>(agg);

        // xw[r] = hin @ W_r  (WMMA f32)
        dim3 gg(N_NODES / 16, N_REL);
        if (P.in_ch == 128)
            k_gemm_xw<128><<<gg, 256, 0, stream>>>(P.hin, W, xw);
        else
            k_gemm_xw<64><<<gg, 256, 0, stream>>>(P.hin, W, xw);

        // attention logits + segment max
        k_logits<<<N_EDGES / 8, 256, 0, stream>>>(xw, src, dst, etype,
                                                  P.q, P.k, logits, mbuf);
        // exp-weights + segment sum + weighted message scatter
        k_scatter<<<N_EDGES / 8, 256, 0, stream>>>(xw, src, dst, etype,
                                                   logits, mbuf, sbuf, agg);
        // normalize, head-mean, bias, skip
        k_combine<<<(N_NODES * OUT_CH + 255) / 256, 256, 0, stream>>>(
            agg, sbuf, P.bias, P.hin, P.in_ch, P.skipW, P.skipb, P.hout);

        if (P.bn)
            k_bn_relu<<<OUT_CH, 256, 0, stream>>>(P.hout, P.bng, P.bnb);
    }
}